// MultivariatePowerFrequencyFeatures_781684048073
// MI455X (gfx1250) — compile-verified
//
#include <hip/hip_runtime.h>
#include <math.h>

// ---------------------------------------------------------------------------
// MultivariatePowerFrequencyFeatures for MI455X (gfx1250, wave32, WMMA)
//
// Pipeline:
//   K1 stft_kernel   : windowed rDFT64 -> specR/specI in ws, layout [b][f][c][t]
//   K2 csd_eig_kernel: per (b,w,f) task (one wave32 each):
//                        G = X X^H / 32 via V_WMMA_F32_16X16X4_F32 (32 wmma)
//                        M = Re(G)^2 + Im(G)^2  -> LDS
//                        parallel cyclic Jacobi eigh (10 sweeps)
//                        out = V diag(log l) V^T via WMMA (4 wmma)
//                        scatter-store transposed output [b][f][c][d][w]
// Memory-bound (~330MB HBM total); fp32 end-to-end for eigensolver accuracy.
// ---------------------------------------------------------------------------

typedef __attribute__((ext_vector_type(2))) float v2f;
typedef __attribute__((ext_vector_type(8))) float v8f;

#define NFFT     64
#define NFREQ    33
#define TFRAMES  1249      // (40000-64)/32 + 1
#define TPAD     1252      // pad so row stride is 16B-aligned
#define TSAMP    40000
#define NB       16
#define NC       16
#define NW       305       // (1249-32)/4 + 1
#define WSTEP    4
#define WINS     32        // frames per window
#define FRB      32        // frames per STFT block
#define NCHUNK   40        // ceil(1249/32)
#define NTASK    (NB * NW * NFREQ)   // 161040 = 8 * 20130

// ---------------- Kernel 1: windowed rDFT (STFT) ----------------
__global__ __launch_bounds__(256)
void stft_kernel(const float* __restrict__ x,
                 float* __restrict__ specR,
                 float* __restrict__ specI)
{
    __shared__ float cR[NFREQ * NFFT];     // DFT*window coefs (real)
    __shared__ float cI[NFREQ * NFFT];     // DFT*window coefs (imag)
    __shared__ float smp[FRB * 32 + 32];   // 1056 samples, frames overlap by 32

    const int tid   = threadIdx.x;
    const int sig   = blockIdx.x / NCHUNK;   // 0..255 = b*16+c
    const int chunk = blockIdx.x % NCHUNK;

    // hann(64, nonperiodic): ||w||^2 = 23.625 exactly -> fold 1/||w|| into coefs
    const float invWnorm = 0.20573779f;
    for (int i = tid; i < NFREQ * NFFT; i += 256) {
        int f = i >> 6, k = i & 63;
        float w = 0.5f * (1.0f - __cosf(6.2831853071795864f * (float)k / 63.0f));
        float ang = -6.2831853071795864f * (float)(f * k) / 64.0f;
        float sn, cs;
        __sincosf(ang, &sn, &cs);
        cR[i] = w * cs * invWnorm;
        cI[i] = w * sn * invWnorm;
    }

    const long t0 = (long)chunk * FRB * 32;
    const float* xs = x + (long)sig * TSAMP;
    for (int i = tid; i < FRB * 32 + 32; i += 256) {
        long g = t0 + i;
        smp[i] = (g < TSAMP) ? xs[g] : 0.0f;
    }
    __syncthreads();

    const int b = sig >> 4, c = sig & 15;
    for (int it = tid; it < FRB * NFREQ; it += 256) {
        int fr = it / NFREQ, fq = it % NFREQ;
        int frame = chunk * FRB + fr;
        if (frame >= TFRAMES) continue;
        float sr = 0.f, si = 0.f;
        const float* sp = &smp[fr * 32];
        const float* pr = &cR[fq * NFFT];
        const float* pi = &cI[fq * NFFT];
        #pragma unroll
        for (int k = 0; k < NFFT; ++k) {
            float v = sp[k];
            sr = fmaf(v, pr[k], sr);
            si = fmaf(v, pi[k], si);
        }
        long off = ((long)(b * NFREQ + fq) * NC + c) * TPAD + frame;
        specR[off] = sr;
        specI[off] = si;
    }
}

// round-robin Jacobi pairing: 15 rounds x 8 disjoint pairs covering all (p,q)
__device__ __forceinline__ void pairpq(int r, int k, int& p, int& q)
{
    if (k == 0) { p = 0; q = 1 + (r % 15); }
    else {
        p = 1 + ((r + k) % 15);
        q = 1 + ((r + 15 - k) % 15);
    }
}

// ---------------- Kernel 2: fused CSD + eigh + recompose ----------------
__global__ __launch_bounds__(256)
void csd_eig_kernel(const float* __restrict__ specR,
                    const float* __restrict__ specI,
                    float* __restrict__ out)
{
    __shared__ float sA[8][16 * 17];   // matrix per wave (stride 17: no bank conflicts)
    __shared__ float sV[8][16 * 17];   // eigenvectors per wave
    __shared__ float sC[8][8];         // rotation cos per pair
    __shared__ float sS[8][8];         // rotation sin per pair
    __shared__ float sL[8][16];        // log-eigenvalues

    const int wv = threadIdx.x >> 5;           // wave in block (8 waves)
    const int l  = threadIdx.x & 31;           // lane
    const int task = blockIdx.x * 8 + wv;      // grid sized exactly: NTASK/8 blocks
    const int b   = task / (NW * NFREQ);
    const int rem = task % (NW * NFREQ);
    const int w   = rem / NFREQ;
    const int f   = rem % NFREQ;

    const int m  = l & 15;                     // A-row / B-col / D-col for this lane
    const int kh = l >> 4;                     // half-wave K offset (0 or 1)

    // ---- CSD: G = X X^H via f32 WMMA; A-tile and B-tile share registers ----
    const float* rowR = specR + ((long)(b * NFREQ + f) * NC + m) * TPAD + w * WSTEP;
    const float* rowI = specI + ((long)(b * NFREQ + f) * NC + m) * TPAD + w * WSTEP;

    v8f crr = {}; v8f cii = {}; v8f cir = {}; v8f cri = {};
    #pragma unroll
    for (int k0 = 0; k0 < WINS; k0 += 4) {
        int kb = k0 + 2 * kh;
        v2f ar, ai;
        ar[0] = rowR[kb]; ar[1] = rowR[kb + 1];
        ai[0] = rowI[kb]; ai[1] = rowI[kb + 1];
        crr = __builtin_amdgcn_wmma_f32_16x16x4_f32(false, ar, false, ar, (short)0, crr, false, false);
        cii = __builtin_amdgcn_wmma_f32_16x16x4_f32(false, ai, false, ai, (short)0, cii, false, false);
        cir = __builtin_amdgcn_wmma_f32_16x16x4_f32(false, ai, false, ar, (short)0, cir, false, false);
        cri = __builtin_amdgcn_wmma_f32_16x16x4_f32(false, ar, false, ai, (short)0, cri, false, false);
    }

    // M = Re(G)^2 + Im(G)^2 -> LDS  (D layout: lane col = l&15, row = e + 8*kh)
    #pragma unroll
    for (int e = 0; e < 8; ++e) {
        float gr = (crr[e] + cii[e]) * (1.0f / 32.0f);
        float gi = (cir[e] - cri[e]) * (1.0f / 32.0f);
        sA[wv][(e + 8 * kh) * 17 + m] = gr * gr + gi * gi;
    }
    // V = I
    #pragma unroll
    for (int e = 0; e < 8; ++e) {
        int idx = l * 8 + e;
        int i = idx >> 4, j = idx & 15;
        sV[wv][i * 17 + j] = (i == j) ? 1.0f : 0.0f;
    }
    __syncthreads();

    // ---- parallel cyclic Jacobi: 10 sweeps x 15 rounds x 8 disjoint rotations ----
    for (int sw = 0; sw < 10; ++sw) {
        for (int r = 0; r < 15; ++r) {
            if (l < 8) {
                int p, q; pairpq(r, l, p, q);
                float app = sA[wv][p * 17 + p];
                float aqq = sA[wv][q * 17 + q];
                float apq = sA[wv][p * 17 + q];
                float cc = 1.0f, ss = 0.0f;
                if (fabsf(apq) > 1e-30f) {
                    float tau = (aqq - app) / (2.0f * apq);
                    float t = 1.0f / (fabsf(tau) + sqrtf(1.0f + tau * tau));
                    if (tau < 0.0f) t = -t;
                    cc = rsqrtf(1.0f + t * t);
                    ss = t * cc;
                }
                sC[wv][l] = cc; sS[wv][l] = ss;
            }
            __syncthreads();
            // row stage: A <- J^T A   (pairs disjoint -> each item's RMW independent)
            for (int rep = 0; rep < 4; ++rep) {
                int it = l + rep * 32;
                int kp = it >> 4, j = it & 15;
                int p, q; pairpq(r, kp, p, q);
                float cc = sC[wv][kp], ss = sS[wv][kp];
                float ap = sA[wv][p * 17 + j], aq = sA[wv][q * 17 + j];
                sA[wv][p * 17 + j] = cc * ap - ss * aq;
                sA[wv][q * 17 + j] = ss * ap + cc * aq;
            }
            __syncthreads();
            // col stage: A <- A J ; V <- V J
            for (int rep = 0; rep < 4; ++rep) {
                int it = l + rep * 32;
                int kp = it >> 4, i = it & 15;
                int p, q; pairpq(r, kp, p, q);
                float cc = sC[wv][kp], ss = sS[wv][kp];
                float a0 = sA[wv][i * 17 + p], a1 = sA[wv][i * 17 + q];
                sA[wv][i * 17 + p] = cc * a0 - ss * a1;
                sA[wv][i * 17 + q] = ss * a0 + cc * a1;
                float v0 = sV[wv][i * 17 + p], v1 = sV[wv][i * 17 + q];
                sV[wv][i * 17 + p] = cc * v0 - ss * v1;
                sV[wv][i * 17 + q] = ss * v0 + cc * v1;
            }
            __syncthreads();
        }
    }

    // log-eigenvalues with nan_to_num(nan=0, neginf=0) semantics
    if (l < 16) {
        float lam = sA[wv][l * 17 + l];
        sL[wv][l] = (lam > 0.0f) ? logf(lam) : 0.0f;
    }
    __syncthreads();

    // ---- recompose: out = (V diag(l)) V^T via WMMA (A,B share V reads) ----
    v8f acc = {};
    #pragma unroll
    for (int k0 = 0; k0 < 16; k0 += 4) {
        int kb = k0 + 2 * kh;
        v2f av, bv;
        bv[0] = sV[wv][m * 17 + kb];
        bv[1] = sV[wv][m * 17 + kb + 1];
        av[0] = bv[0] * sL[wv][kb];
        av[1] = bv[1] * sL[wv][kb + 1];
        acc = __builtin_amdgcn_wmma_f32_16x16x4_f32(false, av, false, bv, (short)0, acc, false, false);
    }

    // scatter-store transposed output: out[b][f][row][col][w]
    #pragma unroll
    for (int e = 0; e < 8; ++e) {
        int mr = e + 8 * kh;
        long off = (((long)(b * NFREQ + f) * 16 + mr) * 16 + m) * NW + w;
        out[off] = acc[e];
    }
}

extern "C" void kernel_launch(void* const* d_in, const int* in_sizes, int n_in,
                              void* d_out, int out_size, void* d_ws, size_t ws_size,
                              hipStream_t stream)
{
    const float* x = (const float*)d_in[0];
    float* specR = (float*)d_ws;                                   // 42.3 MB
    float* specI = specR + (size_t)NB * NFREQ * NC * TPAD;         // 42.3 MB
    float* out   = (float*)d_out;

    stft_kernel<<<dim3(256 * NCHUNK), dim3(256), 0, stream>>>(x, specR, specI);
    csd_eig_kernel<<<dim3(NTASK / 8), dim3(256), 0, stream>>>(specR, specI, out);
}